// GraphNet_41042707480591
// MI455X (gfx1250) — compile-verified
//
#include <hip/hip_runtime.h>
#include <hip/hip_bf16.h>

// ---------------------------------------------------------------------------
// GraphNet on MI455X (gfx1250, wave32):
//   1) h = concat(num_x, cat_emb) @ conv_W.T            (memory-bound)
//   2) deg scatter + rsqrt                              (L2 f32 atomics)
//   3) edge aggregation scatter (norm * h[row] -> agg)  (L2 f32 atomics)
//   4) relu(agg + d^-1 h + b) -> flat                   (streaming)
//   5) g = fc_W[:,20:] @ flat  (shared across batch!)   (streams 16MB of fc_W)
//   6) out = vanilla @ fc_W[:,:20].T + g + fc_b         (V_WMMA_F32_16X16X4_F32)
// ---------------------------------------------------------------------------

#define CONT_LEN     100000
#define N_CAT        8
#define MAX_CAT      32
#define NF           16
#define NCLS         10
#define N_NODES      (CONT_LEN + N_CAT)        // 100008
#define N_EDGES      1600000
#define BATCH        128
#define FLAT_LEN     (N_NODES * 4)             // 400032
#define FC_IN        (FLAT_LEN + 2 * NCLS)     // 400052

typedef __attribute__((ext_vector_type(2))) float v2f;
typedef __attribute__((ext_vector_type(8))) float v8f;

// ---------------- 0) zero workspace (agg | deg | g are contiguous) ----------
__global__ void zero_kernel(float* __restrict__ p, int n) {
    int i = blockIdx.x * blockDim.x + threadIdx.x;
    if (i < n) p[i] = 0.0f;
}

// ---------------- 1) h[i][k] = x[i] . conv_W[k]  (x = num_x or cat_emb) -----
__global__ void h_kernel(const float* __restrict__ num_x,
                         const float* __restrict__ cat_x,
                         const float* __restrict__ embW,
                         const float* __restrict__ embb,
                         const float* __restrict__ convW,
                         float* __restrict__ h) {
    int i = blockIdx.x * blockDim.x + threadIdx.x;
    if (i >= N_NODES) return;
    float xf[NF];
    if (i < CONT_LEN) {
        #pragma unroll
        for (int f = 0; f < NF; ++f) xf[f] = num_x[i * NF + f];
    } else {
        int c = i - CONT_LEN;
        #pragma unroll
        for (int f = 0; f < NF; ++f) {
            float s = embb[c * NF + f];
            #pragma unroll
            for (int m = 0; m < MAX_CAT; ++m)
                s = fmaf(cat_x[c * MAX_CAT + m], embW[(c * NF + f) * MAX_CAT + m], s);
            xf[f] = s;
        }
    }
    #pragma unroll
    for (int k = 0; k < 4; ++k) {
        float s = 0.0f;
        #pragma unroll
        for (int f = 0; f < NF; ++f) s = fmaf(xf[f], convW[k * NF + f], s);
        h[i * 4 + k] = s;
    }
}

// ---------------- 2a) deg[col] += ew ----------------------------------------
__global__ void deg_kernel(const int* __restrict__ ei,
                           const float* __restrict__ ew,
                           float* __restrict__ deg) {
    int e = blockIdx.x * blockDim.x + threadIdx.x;
    if (e >= N_EDGES) return;
    atomicAdd(&deg[ei[N_EDGES + e]], ew[e]);
}

// ---------------- 2b) dinv[i] = rsqrt(deg[i] + 1)  (in place) ---------------
__global__ void rsqrt_kernel(float* __restrict__ deg) {
    int i = blockIdx.x * blockDim.x + threadIdx.x;
    if (i >= N_NODES) return;
    deg[i] = rsqrtf(deg[i] + 1.0f);
}

// ---------------- 3) agg[col] += dinv[row]*ew*dinv[col] * h[row] ------------
__global__ void agg_kernel(const int* __restrict__ ei,
                           const float* __restrict__ ew,
                           const float* __restrict__ dinv,
                           const float* __restrict__ h,
                           float* __restrict__ agg) {
    int e = blockIdx.x * blockDim.x + threadIdx.x;
    if (e >= N_EDGES) return;
    int r = ei[e];
    int c = ei[N_EDGES + e];
    float nrm = dinv[r] * ew[e] * dinv[c];
    const float4 hv = *reinterpret_cast<const float4*>(h + (size_t)r * 4);
    atomicAdd(&agg[c * 4 + 0], nrm * hv.x);
    atomicAdd(&agg[c * 4 + 1], nrm * hv.y);
    atomicAdd(&agg[c * 4 + 2], nrm * hv.z);
    atomicAdd(&agg[c * 4 + 3], nrm * hv.w);
}

// ---------------- 4) flat = relu(agg + dinv^2 * h + conv_b)  (in place) -----
__global__ void fin_kernel(const float* __restrict__ dinv,
                           const float* __restrict__ h,
                           const float* __restrict__ convb,
                           float* __restrict__ agg) {
    int i = blockIdx.x * blockDim.x + threadIdx.x;
    if (i >= N_NODES) return;
    float dd = dinv[i] * dinv[i];
    #pragma unroll
    for (int k = 0; k < 4; ++k) {
        float v = agg[i * 4 + k] + dd * h[i * 4 + k] + convb[k];
        agg[i * 4 + k] = fmaxf(v, 0.0f);
    }
}

// ---------------- 5) g[j] = sum_t fc_W[j][20+t] * flat[t] -------------------
// Streams 16MB of fc_W exactly once (HBM-bound); wave32 shuffle reduction,
// one atomic per wave per class. global_prefetch_b8 keeps the streams ahead.
__global__ void matvec_kernel(const float* __restrict__ fcW,
                              const float* __restrict__ flat,
                              float* __restrict__ g) {
    int tid = blockIdx.x * blockDim.x + threadIdx.x;
    int stride = gridDim.x * blockDim.x;
    float acc[NCLS];
    #pragma unroll
    for (int j = 0; j < NCLS; ++j) acc[j] = 0.0f;
    for (int t = tid; t < FLAT_LEN; t += stride) {
        float fv = flat[t];
        __builtin_prefetch(&flat[t + stride], 0, 1);  // -> global_prefetch_b8
        #pragma unroll
        for (int j = 0; j < NCLS; ++j)
            acc[j] = fmaf(fcW[(size_t)j * FC_IN + 2 * NCLS + t], fv, acc[j]);
    }
    #pragma unroll
    for (int j = 0; j < NCLS; ++j) {
        float v = acc[j];
        #pragma unroll
        for (int off = 16; off > 0; off >>= 1) v += __shfl_down(v, off, 32);
        if ((threadIdx.x & 31) == 0) atomicAdd(&g[j], v);
    }
}

// ---------------- 6) out = vanilla @ fc_W[:, :20].T + g + fc_b --------------
// Single wave32, exact-f32 WMMA: M=128 (8 tiles), N=16 (10 live), K=20 (5x4).
// ISA layouts: A 16x4 f32 -> 2 VGPRs (lanes 0-15: K=0/1, lanes 16-31: K=2/3);
// B 4x16 mirrors it; C/D 16x16 f32 -> 8 VGPRs (VGPR v: M=v / v+8).
__global__ void __launch_bounds__(32) fc_out_kernel(
        const float* __restrict__ vanilla,  // [128, 20]
        const float* __restrict__ fcW,      // [10, FC_IN]
        const float* __restrict__ fcb,      // [10]
        const float* __restrict__ g,        // [10]
        float* __restrict__ out) {          // [128, 10]
    int lane = threadIdx.x;      // 0..31
    int half = lane >> 4;        // 0 | 1
    int l16  = lane & 15;

    // Preload B tiles: B[k][n] = fc_W[n][k], n = l16, k = kc*4 + 2*half + {0,1}
    v2f bt[5];
    #pragma unroll
    for (int kc = 0; kc < 5; ++kc) {
        int kb = kc * 4 + 2 * half;
        float b0 = (l16 < NCLS) ? fcW[(size_t)l16 * FC_IN + kb]     : 0.0f;
        float b1 = (l16 < NCLS) ? fcW[(size_t)l16 * FC_IN + kb + 1] : 0.0f;
        bt[kc].x = b0;
        bt[kc].y = b1;
    }
    float gb = (l16 < NCLS) ? (g[l16] + fcb[l16]) : 0.0f;

    for (int mt = 0; mt < 8; ++mt) {
        int row = mt * 16 + l16;             // A: m = l16
        v8f c = {0.f, 0.f, 0.f, 0.f, 0.f, 0.f, 0.f, 0.f};
        #pragma unroll
        for (int kc = 0; kc < 5; ++kc) {
            int kb = kc * 4 + 2 * half;
            v2f a;
            a.x = vanilla[row * 20 + kb];
            a.y = vanilla[row * 20 + kb + 1];
            c = __builtin_amdgcn_wmma_f32_16x16x4_f32(
                    /*neg_a=*/false, a, /*neg_b=*/false, bt[kc],
                    /*c_mod=*/(short)0, c, /*reuse_a=*/false, /*reuse_b=*/false);
        }
        // D: VGPR v holds (M = mt*16 + v + 8*half, N = l16)
        if (l16 < NCLS) {
            #pragma unroll
            for (int v = 0; v < 8; ++v) {
                int m = mt * 16 + v + 8 * half;
                out[m * NCLS + l16] = c[v] + gb;
            }
        }
    }
}

// ---------------------------------------------------------------------------
extern "C" void kernel_launch(void* const* d_in, const int* in_sizes, int n_in,
                              void* d_out, int out_size, void* d_ws, size_t ws_size,
                              hipStream_t stream) {
    const float* num_x   = (const float*)d_in[0];   // [100000,16]
    const float* cat_x   = (const float*)d_in[1];   // [8,32]
    const int*   ei      = (const int*)  d_in[2];   // [2,1600000]
    const float* ew      = (const float*)d_in[3];   // [1600000]
    const float* vanilla = (const float*)d_in[4];   // [128,20]
    const float* embW    = (const float*)d_in[5];   // [8,16,32]
    const float* embb    = (const float*)d_in[6];   // [8,16]
    const float* convW   = (const float*)d_in[7];   // [4,16]
    const float* convb   = (const float*)d_in[8];   // [4]
    const float* fcW     = (const float*)d_in[9];   // [10,400052]
    const float* fcb     = (const float*)d_in[10];  // [10]
    float* out = (float*)d_out;                     // [128,10]

    float* ws  = (float*)d_ws;
    float* agg = ws;                       // 400032 floats (becomes `flat`)
    float* deg = ws + FLAT_LEN;            // 100008 floats (becomes d^-1/2)
    float* g   = ws + FLAT_LEN + N_NODES;  // 16 floats
    float* h   = ws + FLAT_LEN + N_NODES + 24;  // 400032 floats, 16B aligned

    const int ZN = FLAT_LEN + N_NODES + 16;   // agg + deg + g

    zero_kernel <<<(ZN + 255) / 256, 256, 0, stream>>>(ws, ZN);
    h_kernel    <<<(N_NODES + 127) / 128, 128, 0, stream>>>(num_x, cat_x, embW, embb, convW, h);
    deg_kernel  <<<(N_EDGES + 255) / 256, 256, 0, stream>>>(ei, ew, deg);
    rsqrt_kernel<<<(N_NODES + 255) / 256, 256, 0, stream>>>(deg);
    agg_kernel  <<<(N_EDGES + 255) / 256, 256, 0, stream>>>(ei, ew, deg, h, agg);
    fin_kernel  <<<(N_NODES + 255) / 256, 256, 0, stream>>>(deg, h, convb, agg);
    matvec_kernel<<<256, 256, 0, stream>>>(fcW, agg, g);
    fc_out_kernel<<<1, 32, 0, stream>>>(vanilla, fcW, fcb, g, out);
}